// Model_28054726377946
// MI455X (gfx1250) — compile-verified
//
#include <hip/hip_runtime.h>
#include <hip/hip_bf16.h>

typedef float v2f __attribute__((ext_vector_type(2)));
typedef float v8f __attribute__((ext_vector_type(8)));

#define IMG 256
#define NV 4096
#define NF 1024
#define INV_SIGMA (1.0f / 3e-05f)

// ---------------- K0: zero the scalar loss output ----------------
__global__ void zero_out_kernel(float* out) {
    if (threadIdx.x == 0) out[0] = 0.0f;
}

// ---------------- K1: project vertices -> 2D NDC coords ----------------
// v[j] = sum_k vert[k] * R[j][k] + t[j]; then pinhole + NDC mapping.
__global__ void project_kernel(const float* __restrict__ verts,
                               const float* __restrict__ Kc,
                               const float* __restrict__ R,
                               const float* __restrict__ t,
                               float* __restrict__ v2) {
    int n = blockIdx.x * blockDim.x + threadIdx.x;
    if (n >= NV) return;
    float vx = verts[n * 3 + 0];
    float vy = verts[n * 3 + 1];
    float vz = verts[n * 3 + 2];
    float p0 = vx * R[0] + vy * R[1] + vz * R[2] + t[0];
    float p1 = vx * R[3] + vy * R[4] + vz * R[5] + t[1];
    float p2 = vx * R[6] + vy * R[7] + vz * R[8] + t[2];
    float iz = 1.0f / (p2 + 1e-05f);
    float xn = p0 * iz;
    float yn = p1 * iz;
    float u  = xn * Kc[0] + yn * Kc[1] + Kc[2];
    float vv = xn * Kc[3] + yn * Kc[4] + Kc[5];
    vv = (float)IMG - vv;
    u  = (u  - 128.0f) * (1.0f / 128.0f);
    vv = (vv - 128.0f) * (1.0f / 128.0f);
    v2[n * 2 + 0] = u;
    v2[n * 2 + 1] = vv;
}

// ---------------- K2: per-face edge-function coefficients ----------------
// w_e(px,py) = a_e*px + b_e*py + c_e  (area sign folded in).
// Layout: coeff[(e*3 + comp)*NF + f], comp in {a,b,c}.
__global__ void face_coeff_kernel(const float* __restrict__ v2,
                                  const int* __restrict__ faces,
                                  float* __restrict__ coeff) {
    int f = blockIdx.x * blockDim.x + threadIdx.x;
    if (f >= NF) return;
    int i0 = faces[f * 3 + 0];
    int i1 = faces[f * 3 + 1];
    int i2 = faces[f * 3 + 2];
    float x0 = v2[i0 * 2], y0 = v2[i0 * 2 + 1];
    float x1 = v2[i1 * 2], y1 = v2[i1 * 2 + 1];
    float x2 = v2[i2 * 2], y2 = v2[i2 * 2 + 1];
    float area = (x1 - x0) * (y2 - y0) - (y1 - y0) * (x2 - x0);
    float s = (area > 0.0f) ? 1.0f : ((area < 0.0f) ? -1.0f : 0.0f);
    // edge 0: v0 -> v1
    coeff[0 * NF + f] = -s * (y1 - y0);
    coeff[1 * NF + f] =  s * (x1 - x0);
    coeff[2 * NF + f] =  s * ((y1 - y0) * x0 - (x1 - x0) * y0);
    // edge 1: v1 -> v2
    coeff[3 * NF + f] = -s * (y2 - y1);
    coeff[4 * NF + f] =  s * (x2 - x1);
    coeff[5 * NF + f] =  s * ((y2 - y1) * x1 - (x2 - x1) * y1);
    // edge 2: v2 -> v0
    coeff[6 * NF + f] = -s * (y0 - y2);
    coeff[7 * NF + f] =  s * (x0 - x2);
    coeff[8 * NF + f] =  s * ((y0 - y2) * x2 - (x0 - x2) * y2);
}

// ---------------- K3: WMMA rasterization + loss ----------------
// One wave = 16 consecutive pixels (M). Face loop in chunks of 16 (N).
// A (16x4, f32): K0=px K1=py (lanes 0-15), K2=1 K3=0 (lanes 16-31).
// B_e (4x16, f32): K0=a K1=b (lanes 0-15), K2=c K3=0 (lanes 16-31).
// cov = 1 - PROD_f (1 - sigmoid(d_f/sigma) + 1e-12)  [== 1 - exp(sum log1p)]
__global__ __launch_bounds__(256) void raster_kernel(const float* __restrict__ coeff,
                                                     const float* __restrict__ ref,
                                                     float* __restrict__ out) {
    __shared__ float lds_coeff[9 * NF];
    __shared__ float lds_part[16];

    int tid = threadIdx.x;
    for (int i = tid; i < 9 * NF; i += 256) lds_coeff[i] = coeff[i];
    __syncthreads();

    int lane = tid & 31;
    int waveInBlock = tid >> 5;
    int waveGlobal = blockIdx.x * 8 + waveInBlock;
    int P = waveGlobal * 16;            // base pixel of this wave's 16-pixel tile
    int m = lane & 15;                  // M index within tile / N index within chunk
    int hi = lane >> 4;                 // 0: lanes 0-15 (K=0,1), 1: lanes 16-31 (K=2,3)

    int pixel = P + m;
    int col = pixel & (IMG - 1);
    int row = pixel >> 8;
    float px = (2.0f * (float)col + 1.0f - (float)IMG) * (1.0f / (float)IMG);
    float py = -(2.0f * (float)row + 1.0f - (float)IMG) * (1.0f / (float)IMG);

    v2f A;
    A.x = hi ? 1.0f : px;   // K=0 -> px ; K=2 -> 1
    A.y = hi ? 0.0f : py;   // K=1 -> py ; K=3 -> 0

    // Per-lane PRODUCT accumulators (8 M-rows each); product over this
    // lane's N-columns of (1 - prob + 1e-12).
    v8f acc = {1.0f, 1.0f, 1.0f, 1.0f, 1.0f, 1.0f, 1.0f, 1.0f};

    for (int chunk = 0; chunk < NF / 16; ++chunk) {
        int f = chunk * 16 + m;
        v8f w[3];
#pragma unroll
        for (int e = 0; e < 3; ++e) {
            const float* base = &lds_coeff[e * 3 * NF];
            v2f B;
            B.x = hi ? base[2 * NF + f] : base[0 * NF + f];  // K0: a | K2: c
            B.y = hi ? 0.0f             : base[1 * NF + f];  // K1: b | K3: 0
            v8f cz = {};
            w[e] = __builtin_amdgcn_wmma_f32_16x16x4_f32(
                false, A, false, B, (short)0, cz, false, false);
        }
#pragma unroll
        for (int r = 0; r < 8; ++r) {
            float d = fminf(fminf(w[0][r], w[1][r]), w[2][r]);
            // 1 - sigmoid(x) + 1e-12 = 1/(1+e^x) + 1e-12
            float ex = __expf(d * INV_SIGMA);
            float term = __builtin_amdgcn_rcpf(1.0f + ex) + 1e-12f;
            acc[r] *= term;
        }
    }

    // Reduce over faces (N): product across each 16-lane group.
#pragma unroll
    for (int mask = 1; mask < 16; mask <<= 1) {
#pragma unroll
        for (int r = 0; r < 8; ++r)
            acc[r] *= __shfl_xor(acc[r], mask, 32);
    }

    // Lanes 0 and 16 hold products for M = r + 8*hi. Compute loss partials.
    if (m == 0) {
        float local = 0.0f;
#pragma unroll
        for (int r = 0; r < 8; ++r) {
            int M = r + 8 * hi;
            float cov = 1.0f - acc[r];
            float diff = cov - ref[P + M];
            local += diff * diff;
        }
        lds_part[waveInBlock * 2 + hi] = local;
    }
    __syncthreads();
    if (tid == 0) {
        float total = 0.0f;
#pragma unroll
        for (int i = 0; i < 16; ++i) total += lds_part[i];
        atomicAdd(out, total);
    }
}

extern "C" void kernel_launch(void* const* d_in, const int* in_sizes, int n_in,
                              void* d_out, int out_size, void* d_ws, size_t ws_size,
                              hipStream_t stream) {
    const float* vertices  = (const float*)d_in[0];  // [1,4096,3]
    const float* Kc        = (const float*)d_in[1];  // [1,3,3]
    const float* R         = (const float*)d_in[2];  // [1,3,3]
    const float* t         = (const float*)d_in[3];  // [1,3]
    const float* image_ref = (const float*)d_in[4];  // [256,256]
    const int*   faces     = (const int*)d_in[5];    // [1,1024,3]
    float* out = (float*)d_out;

    float* v2    = (float*)d_ws;        // 4096*2 floats
    float* coeff = v2 + NV * 2;         // 9*1024 floats

    zero_out_kernel<<<1, 32, 0, stream>>>(out);
    project_kernel<<<NV / 256, 256, 0, stream>>>(vertices, Kc, R, t, v2);
    face_coeff_kernel<<<NF / 256, 256, 0, stream>>>(v2, faces, coeff);
    raster_kernel<<<(IMG * IMG) / (16 * 8), 256, 0, stream>>>(coeff, image_ref, out);
}